// WorkingMemory_60662118089466
// MI455X (gfx1250) — compile-verified
//
#include <hip/hip_runtime.h>
#include <hip/hip_bf16.h>
#include <math.h>

#define DELTA 1e-6f
#define B_ 256
#define M_ 512
#define W_ 32
#define R_ 4
#define IN_ 1024
#define TOT_ 247
#define ZLD_ 256

typedef float v2f __attribute__((ext_vector_type(2)));
typedef float v8f __attribute__((ext_vector_type(8)));

__device__ __forceinline__ float softplusf(float x) {
  return (x > 20.f) ? x : log1pf(expf(x));
}
__device__ __forceinline__ float sigmoidf_(float x) {
  return 1.f / (1.f + expf(-x));
}
__device__ __forceinline__ float waveMax(float v) {
#pragma unroll
  for (int o = 16; o > 0; o >>= 1) v = fmaxf(v, __shfl_xor(v, o, 32));
  return v;
}
__device__ __forceinline__ float waveSum(float v) {
#pragma unroll
  for (int o = 16; o > 0; o >>= 1) v += __shfl_xor(v, o, 32);
  return v;
}
// 512-thread block reductions: wave shuffle + 16-partial LDS combine (2 barriers)
__device__ __forceinline__ float blockMax512(float v, float* s_part,
                                             float* s_bc, int t) {
  float w = waveMax(v);
  if ((t & 31) == 0) s_part[t >> 5] = w;
  __syncthreads();
  if (t < 32) {
    float x = (t < 16) ? s_part[t] : -INFINITY;
    x = waveMax(x);
    if (t == 0) *s_bc = x;
  }
  __syncthreads();
  return *s_bc;
}
__device__ __forceinline__ float blockSum512(float v, float* s_part,
                                             float* s_bc, int t) {
  float w = waveSum(v);
  if ((t & 31) == 0) s_part[t >> 5] = w;
  __syncthreads();
  if (t < 32) {
    float x = (t < 16) ? s_part[t] : 0.f;
    x = waveSum(x);
    if (t == 0) *s_bc = x;
  }
  __syncthreads();
  return *s_bc;
}

// ---------------------------------------------------------------------------
// Kernel 1: z = xi @ W_xi + b_xi    (256x1024 @ 1024x247, f32 WMMA 16x16x4)
// One wave per 16x16 tile. N padded to 256: loads use a CLAMPED column (the
// garbage feeds only D columns >= 247, which are never stored) so the inner
// loop has zero predication / EXEC manipulation.
// ---------------------------------------------------------------------------
__global__ void k_gemm_z(const float* __restrict__ xi,
                         const float* __restrict__ Wxi,
                         const float* __restrict__ bxi,
                         float* __restrict__ z) {
  const int tid  = threadIdx.x;
  const int wave = tid >> 5, lane = tid & 31;
  const int half = lane >> 4, lm = lane & 15;
  const int tile = blockIdx.x * 8 + wave;
  const int m0 = (tile >> 4) * 16, n0 = (tile & 15) * 16;
  const int ncol = n0 + lm;
  const bool nok = (ncol < TOT_);
  const int ncs = nok ? ncol : (TOT_ - 1);  // clamped, unconditional loads
  const float* __restrict__ xrow = xi + (size_t)(m0 + lm) * IN_ + 2 * half;
  const float* __restrict__ wcol = Wxi + (size_t)2 * half * TOT_ + ncs;
  v8f acc = {};
  for (int k = 0; k < IN_; k += 4) {
    v2f a, b;
    a.x = xrow[k];
    a.y = xrow[k + 1];
    b.x = wcol[(size_t)k * TOT_];
    b.y = wcol[(size_t)(k + 1) * TOT_];
    acc = __builtin_amdgcn_wmma_f32_16x16x4_f32(false, a, false, b,
                                                (short)0, acc, false, false);
  }
  if (nok) {
    const float bias = bxi[ncol];
#pragma unroll
    for (int j = 0; j < 8; ++j)
      z[(m0 + j + 8 * half) * ZLD_ + ncol] = acc[j] + bias;
  }
}

// ---------------------------------------------------------------------------
// Kernel 2: per-batch state update. One 512-thread block per batch.
// ---------------------------------------------------------------------------
__global__ void k_state(const float* __restrict__ z,
                        const float* __restrict__ memory,
                        const float* __restrict__ read_w,
                        const float* __restrict__ write_w,
                        const float* __restrict__ usage_v,
                        float* __restrict__ ww_out,
                        float* __restrict__ mem_out,
                        float* __restrict__ c_out,
                        float* __restrict__ modes_out) {
  const int b = blockIdx.x;
  const int t = threadIdx.x;  // 0..511
  const float* zb = z + b * ZLD_;
  __shared__ float s_er[W_], s_wv[W_], s_wkn[W_], s_rkn[W_ * R_];
  __shared__ float s_fg[R_], s_rs[R_];
  __shared__ float s_misc[8];  // 0:wstr_eff 1:agate 2:wgate 3:wknorm 4..7:rknorm
  __shared__ float s_red[M_];
  __shared__ float s_u[M_];
  __shared__ int s_idx[M_];
  __shared__ float s_part[16];
  __shared__ float s_bc;

  if (t < W_) {
    s_er[t]  = sigmoidf_(zb[165 + t]);
    s_wv[t]  = zb[197 + t];
    s_wkn[t] = zb[132 + t];
  }
  if (t < R_) {
    s_fg[t] = sigmoidf_(zb[229 + t]);
    s_rs[t] = softplusf(1.f + softplusf(zb[128 + t]));  // softplus(oneplus)
    float a0 = zb[235 + 0 * R_ + t], a1 = zb[235 + 1 * R_ + t],
          a2 = zb[235 + 2 * R_ + t];
    float mx = fmaxf(a0, fmaxf(a1, a2));
    float e0 = expf(a0 - mx), e1 = expf(a1 - mx), e2 = expf(a2 - mx);
    float inv = 1.f / (e0 + e1 + e2);
    modes_out[b * 12 + 0 * R_ + t] = e0 * inv;
    modes_out[b * 12 + 1 * R_ + t] = e1 * inv;
    modes_out[b * 12 + 2 * R_ + t] = e2 * inv;
    float s = 0.f;
    for (int w = 0; w < W_; ++w) { float v = zb[w * R_ + t]; s += v * v; }
    s_misc[4 + t] = sqrtf(s);
  }
  if (t == 0) {
    s_misc[0] = softplusf(1.f + softplusf(zb[164]));
    s_misc[1] = sigmoidf_(zb[233]);
    s_misc[2] = sigmoidf_(zb[234]);
    float s = 0.f;
    for (int w = 0; w < W_; ++w) { float v = zb[132 + w]; s += v * v; }
    s_misc[3] = sqrtf(s);
  }
  __syncthreads();
  if (t < W_) s_wkn[t] = s_wkn[t] / (s_misc[3] + DELTA);
  if (t < W_ * R_) {
    int w = t >> 2, r = t & 3;
    s_rkn[t] = zb[w * R_ + r] / (s_misc[4 + r] + DELTA);
  }
  __syncthreads();

  const int m = t;
  // ---- write content score over OLD memory ----
  const float* mrow = memory + ((size_t)b * M_ + m) * W_;
  float n2 = 0.f, dw = 0.f;
#pragma unroll
  for (int w4 = 0; w4 < W_; w4 += 4) {
    float4 v = *(const float4*)&mrow[w4];
    n2 += v.x * v.x + v.y * v.y + v.z * v.z + v.w * v.w;
    dw += v.x * s_wkn[w4] + v.y * s_wkn[w4 + 1] + v.z * s_wkn[w4 + 2] +
          v.w * s_wkn[w4 + 3];
  }
  float dscore = dw / (sqrtf(n2) + DELTA) * s_misc[0];
  float mx = blockMax512(dscore, s_part, &s_bc, t);
  float ex = expf(dscore - mx);
  float wcw = ex / blockSum512(ex, s_part, &s_bc, t);

  // ---- usage update ----
  float psi = 1.f;
#pragma unroll
  for (int r = 0; r < R_; ++r)
    psi *= (1.f - s_fg[r] * read_w[((size_t)b * M_ + m) * R_ + r]);
  float uv = usage_v[b * M_ + m];
  float usage = (uv + (1.f - uv) * write_w[b * M_ + m]) * psi;
  float u = DELTA + (1.f - DELTA) * usage;

  // ---- allocation: bitonic argsort ascending + inclusive product scan ----
  s_u[t] = u; s_idx[t] = t; __syncthreads();
  for (int k = 2; k <= M_; k <<= 1)
    for (int j = k >> 1; j > 0; j >>= 1) {
      int ixj = t ^ j;
      if (ixj > t) {
        bool up = ((t & k) == 0);
        float a = s_u[t], bb = s_u[ixj];
        if ((a > bb) == up) {
          s_u[t] = bb; s_u[ixj] = a;
          int ia = s_idx[t]; s_idx[t] = s_idx[ixj]; s_idx[ixj] = ia;
        }
      }
      __syncthreads();
    }
  s_red[t] = s_u[t]; __syncthreads();
  for (int off = 1; off < M_; off <<= 1) {
    float v = s_red[t];
    if (t >= off) v *= s_red[t - off];
    __syncthreads();
    s_red[t] = v; __syncthreads();
  }
  float sa = (1.f - s_u[t]) * s_red[t];
  __syncthreads();
  s_u[s_idx[t]] = sa;  // scatter = inverse permutation
  __syncthreads();
  float alloc = s_u[t];

  float agate = s_misc[1], wgate = s_misc[2];
  float ww = wgate * (agate * alloc + (1.f - agate) * wcw);
  ww_out[b * M_ + m] = ww;

  // ---- memory erase/write + read-content dots on NEW memory ----
  float dot[R_] = {0.f, 0.f, 0.f, 0.f};
  float n2n = 0.f;
  float* mo = mem_out + ((size_t)b * M_ + m) * W_;
#pragma unroll
  for (int w4 = 0; w4 < W_; w4 += 4) {
    float4 v = *(const float4*)&mrow[w4];
    float4 o;
    o.x = v.x * (1.f - ww * s_er[w4])     + ww * s_wv[w4];
    o.y = v.y * (1.f - ww * s_er[w4 + 1]) + ww * s_wv[w4 + 1];
    o.z = v.z * (1.f - ww * s_er[w4 + 2]) + ww * s_wv[w4 + 2];
    o.w = v.w * (1.f - ww * s_er[w4 + 3]) + ww * s_wv[w4 + 3];
    *(float4*)&mo[w4] = o;
    n2n += o.x * o.x + o.y * o.y + o.z * o.z + o.w * o.w;
#pragma unroll
    for (int r = 0; r < R_; ++r)
      dot[r] += o.x * s_rkn[w4 * R_ + r] + o.y * s_rkn[(w4 + 1) * R_ + r] +
                o.z * s_rkn[(w4 + 2) * R_ + r] + o.w * s_rkn[(w4 + 3) * R_ + r];
  }
  float invn = 1.f / (sqrtf(n2n) + DELTA);
#pragma unroll
  for (int r = 0; r < R_; ++r) {
    float sc = dot[r] * invn * s_rs[r];
    float mxr = blockMax512(sc, s_part, &s_bc, t);
    float er_ = expf(sc - mxr);
    c_out[((size_t)b * M_ + m) * R_ + r] =
        er_ / blockSum512(er_, s_part, &s_bc, t);
  }
}

// ---------------------------------------------------------------------------
// Kernel 3: single-pass fused link + fwd/bwd einsums over `temporal`.
// Each wave owns a 16-row m-strip and a wave-PRIVATE LDS tile: LDS ops within
// a wave are in-order (ISA DScnt), so no block barriers in the n-loop — only
// wave_barrier() compiler fences. Next tile is prefetched into registers
// while WMMAs consume the current tile. rw indexed with (lm&3): duplicate
// values feed only discarded D columns, so no predication anywhere.
// ---------------------------------------------------------------------------
__device__ __forceinline__ float linkval(int m, int n, float tv,
                                         const float* ww, const float* tw) {
  if (m == n) return 0.f;
  const float wm = ww[m];
  return (1.f - wm - ww[n]) * tv + wm * tw[n];
}

__global__ void k_link_fwd_bwd(const float* __restrict__ temporal,
                               const float* __restrict__ read_w,
                               const float* __restrict__ tw_g,
                               const float* __restrict__ ww_g,
                               float* __restrict__ fwd,
                               float* __restrict__ bwd) {
  __shared__ float s_ww[M_], s_tw[M_], s_rw[M_ * R_];
  __shared__ float s_tile[8][16 * 17];  // stride-17: conflict-free both axes
  const int b = blockIdx.y;
  const int t = threadIdx.x;  // 256
  const int wave = t >> 5, lane = t & 31, half = lane >> 4, lm = lane & 15;
  const int m0 = blockIdx.x * 128 + wave * 16;

  for (int i = t; i < M_; i += 256) {
    s_ww[i] = ww_g[b * M_ + i];
    s_tw[i] = tw_g[b * M_ + i];
  }
  for (int i = t; i < M_ * R_; i += 256)
    s_rw[i] = read_w[(size_t)b * M_ * R_ + i];
  __syncthreads();

  const int lr = lm & 3;  // clamped r: lanes 4..15 feed discarded D columns
  float bbx[4], bby[4];
#pragma unroll
  for (int c = 0; c < 4; ++c) {
    const int mr = m0 + 4 * c + 2 * half;
    bbx[c] = s_rw[mr * R_ + lr];
    bby[c] = s_rw[(mr + 1) * R_ + lr];
  }

  // per-lane global/LDS addresses for the two float4 chunks of the tile
  const int r0 = lane >> 2,        c0 = (lane & 3) * 4;
  const int r1 = (lane + 32) >> 2, c1 = (lane & 3) * 4;
  const float* __restrict__ g0 =
      temporal + ((size_t)b * M_ + m0 + r0) * M_ + c0;
  const float* __restrict__ g1 =
      temporal + ((size_t)b * M_ + m0 + r1) * M_ + c1;
  float* tile = s_tile[wave];
  float* d0 = &tile[r0 * 17 + c0];
  float* d1 = &tile[r1 * 17 + c1];

  float4 pa = *(const float4*)g0;
  float4 pb = *(const float4*)g1;
  v8f accf = {};
  for (int nt = 0; nt < 32; ++nt) {
    d0[0] = pa.x; d0[1] = pa.y; d0[2] = pa.z; d0[3] = pa.w;
    d1[0] = pb.x; d1[1] = pb.y; d1[2] = pb.z; d1[3] = pb.w;
    const int ntn = (nt + 1 < 32) ? (nt + 1) : 31;  // harmless re-load on last
    pa = *(const float4*)(g0 + ntn * 16);
    pb = *(const float4*)(g1 + ntn * 16);
    __builtin_amdgcn_wave_barrier();  // reads stay after stores
    const int n0 = nt * 16;

    // fwd: accf += link_tile(16m x 16n) @ rw(16n x 16rp), 4 K=4 chunks
#pragma unroll
    for (int c = 0; c < 4; ++c) {
      const int col0 = 4 * c + 2 * half;
      v2f a, bv;
      a.x = linkval(m0 + lm, n0 + col0,     tile[lm * 17 + col0],     s_ww, s_tw);
      a.y = linkval(m0 + lm, n0 + col0 + 1, tile[lm * 17 + col0 + 1], s_ww, s_tw);
      bv.x = s_rw[(n0 + col0) * R_ + lr];
      bv.y = s_rw[(n0 + col0 + 1) * R_ + lr];
      accf = __builtin_amdgcn_wmma_f32_16x16x4_f32(false, a, false, bv,
                                                   (short)0, accf, false, false);
    }
    // bwd partial: accb = link_tileT(16n x 16m) @ rw_mstrip(16m x 16rp)
    v8f accb = {};
#pragma unroll
    for (int c = 0; c < 4; ++c) {
      const int mc0 = 4 * c + 2 * half;
      v2f a, bv;
      a.x = linkval(m0 + mc0,     n0 + lm, tile[mc0 * 17 + lm],       s_ww, s_tw);
      a.y = linkval(m0 + mc0 + 1, n0 + lm, tile[(mc0 + 1) * 17 + lm], s_ww, s_tw);
      bv.x = bbx[c];
      bv.y = bby[c];
      accb = __builtin_amdgcn_wmma_f32_16x16x4_f32(false, a, false, bv,
                                                   (short)0, accb, false, false);
    }
    if (lm < R_) {
#pragma unroll
      for (int j = 0; j < 8; ++j)
        atomicAdd(&bwd[((size_t)b * M_ + n0 + j + 8 * half) * R_ + lm], accb[j]);
    }
    __builtin_amdgcn_wave_barrier();  // next stores stay after reads
  }
  if (lm < R_) {
#pragma unroll
    for (int j = 0; j < 8; ++j)
      fwd[((size_t)b * M_ + m0 + j + 8 * half) * R_ + lm] = accf[j];
  }
}

// ---------------------------------------------------------------------------
// Kernel 4: rw_final = modes combine; read_vectors = memT @ rw_final per batch
// ---------------------------------------------------------------------------
__global__ void k_read(const float* __restrict__ mem,
                       const float* __restrict__ c,
                       const float* __restrict__ fwd,
                       const float* __restrict__ bwd,
                       const float* __restrict__ modes,
                       float* __restrict__ out) {
  const int b = blockIdx.x, t = threadIdx.x;  // 256
  __shared__ float s_rwf[M_ * R_];
  __shared__ float s_m[12];
  if (t < 12) s_m[t] = modes[b * 12 + t];
  __syncthreads();
  for (int i = t; i < M_ * R_; i += 256) {
    const int r = i & 3;
    const size_t gi = (size_t)b * M_ * R_ + i;
    s_rwf[i] = s_m[0 * R_ + r] * bwd[gi] + s_m[1 * R_ + r] * c[gi] +
               s_m[2 * R_ + r] * fwd[gi];
  }
  __syncthreads();
  if (t < W_ * R_) {
    const int w = t >> 2, r = t & 3;
    float acc = 0.f;
    for (int m = 0; m < M_; ++m)
      acc += mem[((size_t)b * M_ + m) * W_ + w] * s_rwf[m * R_ + r];
    out[b * W_ * R_ + w * R_ + r] = acc;
  }
}

extern "C" void kernel_launch(void* const* d_in, const int* in_sizes, int n_in,
                              void* d_out, int out_size, void* d_ws,
                              size_t ws_size, hipStream_t stream) {
  const float* xi       = (const float*)d_in[0];
  const float* memory   = (const float*)d_in[1];
  const float* temporal = (const float*)d_in[2];
  const float* tw       = (const float*)d_in[3];
  const float* rw       = (const float*)d_in[4];
  const float* wwts     = (const float*)d_in[5];
  const float* uv       = (const float*)d_in[6];
  const float* Wxi      = (const float*)d_in[7];
  const float* bxi      = (const float*)d_in[8];

  float* ws    = (float*)d_ws;
  float* z     = ws;                  // 256*256        = 65536
  float* ww    = z + 65536;           // 256*512        = 131072
  float* mem   = ww + 131072;         // 256*512*32     = 4194304
  float* c     = mem + 4194304;       // 256*512*4      = 524288
  float* fwd   = c + 524288;          // 524288
  float* bwd   = fwd + 524288;        // 524288
  float* modes = bwd + 524288;        // 256*12         = 3072

  hipMemsetAsync(bwd, 0, 524288 * sizeof(float), stream);
  k_gemm_z<<<32, 256, 0, stream>>>(xi, Wxi, bxi, z);
  k_state<<<256, 512, 0, stream>>>(z, memory, rw, wwts, uv, ww, mem, c, modes);
  k_link_fwd_bwd<<<dim3(4, 256), 256, 0, stream>>>(temporal, rw, tw, ww, fwd, bwd);
  k_read<<<256, 256, 0, stream>>>(mem, c, fwd, bwd, modes, (float*)d_out);
}